// FPNRoIAlign_841813590621
// MI455X (gfx1250) — compile-verified
//
#include <hip/hip_runtime.h>
#include <hip/hip_bf16.h>

typedef __attribute__((ext_vector_type(2))) float v2f;
typedef __attribute__((ext_vector_type(8))) float v8f;

#define POOLED   7
#define GRID     14            // POOLED * sampling_ratio(2)
#define NSAMP    196           // 14*14 bilinear samples (cell-major order)
#define KPAD     224           // sample dim padded to 7*32 for lane-strided gather
#define CCHUNK   64            // channels per LDS chunk
#define SSTRIDE  228           // LDS row stride (floats), conflict-free A reads
#define CHANNELS 256
#define CELLS    49            // 7*7 pooled cells

__global__ __launch_bounds__(256) void fpn_roi_align_kernel(
    const float* __restrict__ f0, const float* __restrict__ f1,
    const float* __restrict__ f2, const float* __restrict__ f3,
    const float* __restrict__ rois, float* __restrict__ out, int nRoi)
{
    __shared__ __align__(16) float sS[CCHUNK * SSTRIDE];   // 58368 B
    __shared__ float sFy[GRID], sVy[GRID], sFx[GRID], sVx[GRID];
    __shared__ int   sY0[GRID], sY1[GRID], sX0[GRID], sX1[GRID];

    const int roi  = blockIdx.x;
    if (roi >= nRoi) return;
    const int tid  = threadIdx.x;
    const int lane = tid & 31;

    // wave-uniform values forced into SGPRs (scalar branches + SADDR addressing)
    const int waveU = __builtin_amdgcn_readfirstlane(tid >> 5);
    const int mtU   = waveU & 3;            // 16-channel tile within chunk
    const int ntU   = (waveU >> 2) << 1;    // first of two 16-cell tiles (0 or 2)

    // ---------- uniform per-ROI setup (scalar path) ----------
    const float rb  = rois[roi * 5 + 0];
    const float rx1 = rois[roi * 5 + 1];
    const float ry1 = rois[roi * 5 + 2];
    const float rx2 = rois[roi * 5 + 3];
    const float ry2 = rois[roi * 5 + 4];

    const float ww = rx2 - rx1 + 1.0f;
    const float hh = ry2 - ry1 + 1.0f;
    int lvl = (int)floorf(2.0f + log2f(sqrtf(ww * hh) / 224.0f + 1e-6f));
    lvl = lvl < 0 ? 0 : (lvl > 3 ? 3 : lvl);
    const float scale = 0.25f / (float)(1 << lvl);
    const int   HS    = 256 >> lvl;                 // square feature maps
    const float*feat  = (lvl == 0) ? f0 : (lvl == 1) ? f1 : (lvl == 2) ? f2 : f3;
    const int   bIdx  = (int)rb;

    const float x1s  = rx1 * scale, y1s = ry1 * scale;
    const float roiW = fmaxf(rx2 * scale - x1s, 1.0f);
    const float roiH = fmaxf(ry2 * scale - y1s, 1.0f);
    const float binW = roiW / (float)POOLED;
    const float binH = roiH / (float)POOLED;

    // ---------- separable sample coordinates ----------
    if (tid < GRID) {
        float g  = ((float)tid + 0.5f) * 0.5f;
        float sy = y1s + binH * g;
        float v  = (sy >= -1.0f && sy <= (float)HS) ? 1.0f : 0.0f;
        sy = fminf(fmaxf(sy, 0.0f), (float)HS - 1.0f);
        int lo = (int)floorf(sy); if (lo > HS - 1) lo = HS - 1;
        int hi = lo + 1;          if (hi > HS - 1) hi = HS - 1;
        sY0[tid] = lo; sY1[tid] = hi; sFy[tid] = sy - (float)lo; sVy[tid] = v;
    } else if (tid >= 32 && tid < 32 + GRID) {
        int  t  = tid - 32;
        float g  = ((float)t + 0.5f) * 0.5f;
        float sx = x1s + binW * g;
        float v  = (sx >= -1.0f && sx <= (float)HS) ? 1.0f : 0.0f;
        sx = fminf(fmaxf(sx, 0.0f), (float)HS - 1.0f);
        int lo = (int)floorf(sx); if (lo > HS - 1) lo = HS - 1;
        int hi = lo + 1;          if (hi > HS - 1) hi = HS - 1;
        sX0[t] = lo; sX1[t] = hi; sFx[t] = sx - (float)lo; sVx[t] = v;
    }
    __syncthreads();

    const size_t HW    = (size_t)HS * (size_t)HS;
    const float* fbase = feat + (size_t)bIdx * CHANNELS * HW;
    float* outbase     = out + (size_t)roi * (CHANNELS * CELLS);

    const int n0     = ntU * 16 + (lane & 15);      // always < 49
    const int n1     = n0 + 16;                     // group0: <32; group2: 48..63
    const int chRow  = mtU * 16 + (lane & 15);      // channel row within chunk
    const int kHalf  = (lane >> 4) << 1;            // K offset per lane-half

    for (int cbase = 0; cbase < CHANNELS; cbase += CCHUNK) {
        // ---------- gather + bilinear: S[c_local][s'] -> LDS (cell-major s') ----
        const float* fp0 = fbase + (size_t)(cbase + waveU * 8) * HW;
        #pragma unroll
        for (int i = 0; i < KPAD / 32; ++i) {
            const int s = i * 32 + lane;          // lane-contiguous samples
            int off00 = 0, off01 = 0, off10 = 0, off11 = 0;
            float fx = 0.0f, fy = 0.0f, m = 0.0f;
            if (s < NSAMP) {                      // divergent only in last iter
                const int cell = s >> 2;          // pooled cell (cell-major order)
                const int sub  = s & 3;           // 2x2 sub-sample within cell
                const int ph   = (cell * 4682) >> 15;   // cell / 7, exact <= 48
                const int pw   = cell - ph * 7;
                const int iy   = 2 * ph + (sub >> 1);
                const int ix   = 2 * pw + (sub & 1);
                const int rowA = sY0[iy] * HS, rowB = sY1[iy] * HS;
                const int x0 = sX0[ix], x1i = sX1[ix];
                off00 = rowA + x0; off01 = rowA + x1i;
                off10 = rowB + x0; off11 = rowB + x1i;
                fy = sFy[iy]; fx = sFx[ix];
                m  = sVy[iy] * sVx[ix];
            }
            #pragma unroll
            for (int j = 0; j < 8; ++j) {         // 8 channels reuse the offsets
                const float* fp = fp0 + (size_t)j * HW;
                const float v00 = fp[off00];
                const float v01 = fp[off01];
                const float v10 = fp[off10];
                const float v11 = fp[off11];
                const float top = v00 + fx * (v01 - v00);
                const float bot = v10 + fx * (v11 - v10);
                sS[(waveU * 8 + j) * SSTRIDE + s] = (top + fy * (bot - top)) * m;
            }
        }
        __syncthreads();

        // ---------- WMMA pooling (block-diagonal): tile nt <- kt in [16nt,16nt+16)
        const float* aRow = &sS[chRow * SSTRIDE + kHalf];

        auto mm16 = [&](int ktBase, int n, v8f acc) {
            #pragma unroll
            for (int t = 0; t < 16; ++t) {
                const int kt = ktBase + t;
                v2f a = *(const v2f*)(aRow + kt * 4);
                const float bv = (n == kt) ? 0.25f : 0.0f;
                v2f b = { bv, bv };
                acc = __builtin_amdgcn_wmma_f32_16x16x4_f32(
                          false, a, false, b, (short)0, acc, false, false);
            }
            return acc;
        };

        v8f acc0 = mm16(ntU * 16, n0, v8f{});
        v8f acc1 = {};
        if (ntU == 0) {                          // scalar branch: wave-uniform
            acc1 = mm16(16, n1, acc1);
        } else {                                 // nt=3 tile: only cell 48 exists
            v2f a = *(const v2f*)(aRow + 48 * 4);
            const float bv = (n1 == 48) ? 0.25f : 0.0f;
            v2f b = { bv, bv };
            acc1 = __builtin_amdgcn_wmma_f32_16x16x4_f32(
                       false, a, false, b, (short)0, acc1, false, false);
        }

        // ---------- store D tiles: out[c*49 + cell], contiguous across lanes ----
        const int cRowBase = cbase + mtU * 16 + ((lane >> 4) << 3);
        #pragma unroll
        for (int r = 0; r < 8; ++r) {            // n0 always in range: no guard
            __builtin_nontemporal_store(acc0[r], &outbase[(cRowBase + r) * CELLS + n0]);
        }
        if (ntU == 0) {                          // n1 in [16,32): no guard
            #pragma unroll
            for (int r = 0; r < 8; ++r) {
                __builtin_nontemporal_store(acc1[r], &outbase[(cRowBase + r) * CELLS + n1]);
            }
        } else if ((lane & 15) == 0) {           // only cell 48 of nt=3 is real
            #pragma unroll
            for (int r = 0; r < 8; ++r) {
                __builtin_nontemporal_store(acc1[r], &outbase[(cRowBase + r) * CELLS + 48]);
            }
        }
        __syncthreads();
    }
}

extern "C" void kernel_launch(void* const* d_in, const int* in_sizes, int n_in,
                              void* d_out, int out_size, void* d_ws, size_t ws_size,
                              hipStream_t stream) {
    const float* f0   = (const float*)d_in[0];
    const float* f1   = (const float*)d_in[1];
    const float* f2   = (const float*)d_in[2];
    const float* f3   = (const float*)d_in[3];
    const float* rois = (const float*)d_in[4];
    const int nRoi = in_sizes[4] / 5;
    fpn_roi_align_kernel<<<dim3(nRoi), dim3(256), 0, stream>>>(
        f0, f1, f2, f3, rois, (float*)d_out, nRoi);
}